// AttenFeature_67551245631580
// MI455X (gfx1250) — compile-verified
//
#include <hip/hip_runtime.h>
#include <math.h>

#define B_N   128
#define F_N   2048
#define R_N   49
#define I_N   312
#define I_PAD 320    // 20 * 16
#define V_N   300
#define V3_N  900
#define V3_PAD 912   // 57 * 16

typedef float v2f __attribute__((ext_vector_type(2)));
typedef float v8f __attribute__((ext_vector_type(8)));

__device__ __forceinline__ v8f wmma_f32(v2f a, v2f b, v8f c) {
  // 8 args: (neg_a, A, neg_b, B, c_mod, C, reuse_a, reuse_b)
  return __builtin_amdgcn_wmma_f32_16x16x4_f32(
      false, a, false, b, (short)0, c, false, false);
}

__device__ __forceinline__ int imin(int a, int b) { return a < b ? a : b; }

// ---------------------------------------------------------------------------
// Kernel 1: VW2 = V[312,300] @ W2[300,2048], stored padded to [320,2048] in ws
// (pad rows written as 0 so downstream loads need no guards).
// One wave per 16x16 tile. K = 300 = 75 * 4. A-rows >= 312 are loaded from a
// clamped address (junk) and overwritten with 0 at store time.
// ---------------------------------------------------------------------------
__global__ void vw2_kernel(const float* __restrict__ V,
                           const float* __restrict__ W2,
                           float* __restrict__ vw2) {
  const int lane = threadIdx.x & 31;
  const int lo = lane & 15;
  const int hi = lane >> 4;
  const int f0 = blockIdx.x * 16;
  const int i0 = blockIdx.y * 16;

  const int amc = imin(i0 + lo, I_N - 1);   // clamped A row -> no branches
  const int bn = f0 + lo;                   // B col (N)
  const float* arow = V + (size_t)amc * V_N;

  v8f acc = {};
  #pragma unroll 5
  for (int k = 0; k < V_N; k += 4) {
    const int kb = k + 2 * hi;
    v2f a = *(const v2f*)(arow + kb);       // contiguous, 8B aligned
    v2f b;
    b.x = W2[kb * F_N + bn];
    b.y = W2[(kb + 1) * F_N + bn];
    acc = wmma_f32(a, b, acc);
  }
  #pragma unroll
  for (int v = 0; v < 8; ++v) {
    const int row = i0 + v + 8 * hi;
    vw2[(size_t)row * F_N + f0 + lo] = (row < I_N) ? acc[v] : 0.0f;
  }
}

// ---------------------------------------------------------------------------
// Kernel 2: per (i_tile, b): logits = VW2_tile @ Fsr[b]  -> softmax over r
//           -> AttFs_tile = A_sm @ Fsr[b]^T  (written to output)
// Block = 128 threads (4 waves). VW2 A-tile (128 KB) and logits live in LDS.
// ---------------------------------------------------------------------------
__global__ void attn_kernel(const float* __restrict__ Fs,
                            const float* __restrict__ vw2,
                            float* __restrict__ outA) {
  __shared__ float As[16 * F_N];   // 128 KB: VW2 rows i0..i0+15
  __shared__ float Lg[16][64];     // logits / softmax tile

  const int tid = threadIdx.x;
  const int wave = tid >> 5;
  const int lane = tid & 31;
  const int lo = lane & 15;
  const int hi = lane >> 4;
  const int i0 = blockIdx.x * 16;
  const int b = blockIdx.y;
  const float* fsb = Fs + (size_t)b * F_N * R_N;

  // Stage A tile once, coalesced; all 4 waves share it.
  {
    const float* vrow = vw2 + (size_t)i0 * F_N;
    for (int idx = tid; idx < 16 * F_N; idx += 128) As[idx] = vrow[idx];
  }
  __syncthreads();

  // ---- Step 1: logits[16, 64] (each wave owns one 16-wide r tile) ----
  // Columns r >= 49 use a clamped address (duplicate junk); softmax zeroes
  // them before first use, so no value select is needed here.
  {
    const int r0 = wave * 16;
    const int rnc = imin(r0 + lo, R_N - 1);
    v8f acc = {};
    #pragma unroll 4
    for (int k = 0; k < F_N; k += 4) {
      const int kb = k + 2 * hi;
      v2f a = *(const v2f*)(As + lo * F_N + kb);         // ds_load pair
      v2f bb;
      bb.x = fsb[kb * R_N + rnc];
      bb.y = fsb[(kb + 1) * R_N + rnc];
      acc = wmma_f32(a, bb, acc);
    }
    #pragma unroll
    for (int v = 0; v < 8; ++v)
      Lg[v + 8 * hi][r0 + lo] = acc[v];
  }
  __syncthreads();

  // ---- Step 2: softmax over the 49 valid r per row; zero the pad cols ----
  if (tid < 16) {
    float mx = -3.4e38f;
    for (int r = 0; r < R_N; ++r) mx = fmaxf(mx, Lg[tid][r]);
    float s = 0.0f;
    for (int r = 0; r < R_N; ++r) {
      const float e = __expf(Lg[tid][r] - mx);
      Lg[tid][r] = e;
      s += e;
    }
    const float inv = 1.0f / s;
    for (int r = 0; r < R_N; ++r) Lg[tid][r] *= inv;
    for (int r = R_N; r < 64; ++r) Lg[tid][r] = 0.0f;   // exact zero K-pad
  }
  __syncthreads();

  // ---- Step 3: AttFs_tile[16, 2048] = A_sm[16,64] @ B, B[k=r][n=f] ----
  // A is exactly 0 for k >= 49, so clamped B loads contribute exactly 0.
  for (int t = wave; t < F_N / 16; t += 4) {
    const int f0 = t * 16;
    const int base = (f0 + lo) * R_N;
    v8f acc = {};
    #pragma unroll
    for (int k = 0; k < 64; k += 4) {
      const int kb = k + 2 * hi;
      v2f a;
      a.x = Lg[lo][kb];
      a.y = Lg[lo][kb + 1];
      v2f bb;
      bb.x = fsb[base + imin(kb, R_N - 1)];
      bb.y = fsb[base + imin(kb + 1, R_N - 1)];
      acc = wmma_f32(a, bb, acc);
    }
    #pragma unroll
    for (int v = 0; v < 8; ++v) {
      const int irow = i0 + v + 8 * hi;
      if (irow < I_N)
        outA[((size_t)b * I_N + irow) * F_N + f0 + lo] = acc[v];
    }
  }
}

// ---------------------------------------------------------------------------
// Kernel 3: fused MLP. AttFs viewed as flat [39936, 2048] (contiguous).
// Per block: 16 rows. tmp[16,912] = rows @ lin1_w + b1 kept in LDS, then
// out[16,300] = tmp @ lin2_w + b2. Avoids 144 MB linear_out round-trip.
// LDS: 128 KB A-tile + 57 KB tmp < 320 KB WGP budget.
// ---------------------------------------------------------------------------
__global__ void mlp_kernel(const float* __restrict__ attfs,
                           const float* __restrict__ l1w,
                           const float* __restrict__ l1b,
                           const float* __restrict__ l2w,
                           const float* __restrict__ l2b,
                           float* __restrict__ outV) {
  __shared__ float As[16 * F_N];        // 128 KB
  __shared__ float Tm[16 * V3_PAD];     // 57 KB

  const int tid = threadIdx.x;
  const int wave = tid >> 5;
  const int lane = tid & 31;
  const int lo = lane & 15;
  const int hi = lane >> 4;
  const int row0 = blockIdx.x * 16;

  // Stage A tile (coalesced)
  const float* arow = attfs + (size_t)row0 * F_N;
  for (int idx = tid; idx < 16 * F_N; idx += 128) As[idx] = arow[idx];
  __syncthreads();

  // ---- Phase 1: tmp = A @ lin1_w + b1 ; n-tiles wave-strided ----
  // Columns >= 900 use clamped addresses (junk accumulator) and are stored
  // as exact zeros, which phase 2 relies on.
  for (int t = wave; t < V3_PAD / 16; t += 4) {
    const int n0 = t * 16;
    const int nc = n0 + lo;
    const bool nv = nc < V3_N;
    const int ncc = imin(nc, V3_N - 1);
    v8f acc = {};
    #pragma unroll 4
    for (int k = 0; k < F_N; k += 4) {
      const int kb = k + 2 * hi;
      v2f a = *(const v2f*)(As + lo * F_N + kb);         // ds_load pair
      v2f bb;
      bb.x = l1w[(size_t)kb * V3_N + ncc];
      bb.y = l1w[(size_t)(kb + 1) * V3_N + ncc];
      acc = wmma_f32(a, bb, acc);
    }
    const float bias = l1b[ncc];
    #pragma unroll
    for (int v = 0; v < 8; ++v)
      Tm[(v + 8 * hi) * V3_PAD + n0 + lo] = nv ? (acc[v] + bias) : 0.0f;
  }
  __syncthreads();

  // ---- Phase 2: out = tmp @ lin2_w + b2 ; K padded to 912 ----
  // Tm cols 900..911 are exact zeros => clamped B rows contribute 0.
  for (int t = wave; t < 19; t += 4) {                   // 19*16 = 304 >= 300
    const int n0 = t * 16;
    const int nc = n0 + lo;
    const bool nv = nc < V_N;
    const int ncc = imin(nc, V_N - 1);
    v8f acc = {};
    #pragma unroll 4
    for (int k = 0; k < V3_PAD; k += 4) {
      const int kb = k + 2 * hi;
      v2f a = *(const v2f*)(Tm + lo * V3_PAD + kb);      // ds_load pair
      v2f bb;
      bb.x = l2w[imin(kb, V3_N - 1) * V_N + ncc];
      bb.y = l2w[imin(kb + 1, V3_N - 1) * V_N + ncc];
      acc = wmma_f32(a, bb, acc);
    }
    const float bias = l2b[ncc];
    #pragma unroll
    for (int v = 0; v < 8; ++v) {
      if (nv)
        outV[(size_t)(row0 + v + 8 * hi) * V_N + nc] = acc[v] + bias;
    }
  }
}

// ---------------------------------------------------------------------------
// Kernel 4: Img_fs[b,f] = mean_r Fs[b,f,r]   (L2-resident by now)
// ---------------------------------------------------------------------------
__global__ void imgfs_kernel(const float* __restrict__ Fs,
                             float* __restrict__ outI) {
  const int idx = blockIdx.x * blockDim.x + threadIdx.x;
  if (idx >= B_N * F_N) return;
  const float* p = Fs + (size_t)idx * R_N;
  float s = 0.0f;
  #pragma unroll
  for (int r = 0; r < R_N; ++r) s += p[r];
  outI[idx] = s * (1.0f / (float)R_N);
}

// ---------------------------------------------------------------------------
extern "C" void kernel_launch(void* const* d_in, const int* in_sizes, int n_in,
                              void* d_out, int out_size, void* d_ws, size_t ws_size,
                              hipStream_t stream) {
  const float* Fs  = (const float*)d_in[0];
  const float* V   = (const float*)d_in[1];
  // d_in[2] = W1: only feeds the unused S einsum -> skipped entirely.
  const float* W2  = (const float*)d_in[3];
  const float* l1w = (const float*)d_in[4];
  const float* l1b = (const float*)d_in[5];
  const float* l2w = (const float*)d_in[6];
  const float* l2b = (const float*)d_in[7];

  float* out  = (float*)d_out;
  float* outA = out;                                        // [128,312,2048]
  float* outV = out + (size_t)B_N * I_N * F_N;              // [128,312,300]
  float* outI = outV + (size_t)B_N * I_N * V_N;             // [128,2048]
  float* vw2  = (float*)d_ws;                               // [320,2048] = 2.62 MB

  vw2_kernel<<<dim3(F_N / 16, I_PAD / 16), 32, 0, stream>>>(V, W2, vw2);
  attn_kernel<<<dim3(I_PAD / 16, B_N), 128, 0, stream>>>(Fs, vw2, outA);
  mlp_kernel<<<dim3((B_N * I_N) / 16), 128, 0, stream>>>(outA, l1w, l1b,
                                                         l2w, l2b, outV);
  imgfs_kernel<<<dim3((B_N * F_N + 255) / 256), 256, 0, stream>>>(Fs, outI);
}